// SCAPLinearRealSparse_79611513799417
// MI455X (gfx1250) — compile-verified
//
#include <hip/hip_runtime.h>

// ---------------------------------------------------------------------------
// SCAP sparse-linear decode, folded to a single-pass fp32 GEMV:
//   out = v @ W + bias,  v_i = (|x_i - MODE| > THRESH) ? x_i : MODE
// Memory-bound: W = 939.5 MB streamed once (~40 us floor at 23.3 TB/s).
// - x chunk staged to LDS with GLOBAL_LOAD_ASYNC_TO_LDS_B128 (ASYNCcnt DMA),
//   then demoded in place.
// - Accumulation on V_WMMA_F32_16X16X4_F32 (wave32); B operands are per-lane
//   b32 loads but each half-wave covers one contiguous 64B sector -> full
//   bandwidth efficiency; A operand broadcast via ds_load_b64 from LDS.
// - Deterministic split-K (partials in d_ws) + tiny bias-reduce kernel.
// ---------------------------------------------------------------------------

typedef __attribute__((ext_vector_type(2))) float v2f;
typedef __attribute__((ext_vector_type(8))) float v8f;

#define MODE_C   0.09f
#define THRESH_C 0.1f
#define IN_F     8192
#define OUT_F    28672
#define KSPLIT   4
#define KCHUNK   (IN_F / KSPLIT)      // 2048 floats = 8 KB per block
#define COLS_PER_BLOCK 128            // 8 waves * 16 columns
#define THREADS  256

__global__ __launch_bounds__(THREADS)
void scap_gemv_wmma_kernel(const float* __restrict__ x,
                           const float* __restrict__ W,
                           float* __restrict__ partial) {
    __shared__ float vsh[KCHUNK];

    const int tid = threadIdx.x;
    const int k0  = blockIdx.y * KCHUNK;

    // ---- Stage x[k0 : k0+KCHUNK] into LDS with async DMA (b128 per lane) ----
    // Flat shared-pointer low 32 bits == wave-relative LDS byte offset.
    {
        unsigned lds_base = (unsigned)(size_t)(void*)vsh;
        const float* gsrc = x + k0 + tid * 4;           // 16B-aligned
        #pragma unroll
        for (int it = 0; it < KCHUNK / (THREADS * 4); ++it) {
            unsigned ldst = lds_base + (unsigned)(tid * 16 + it * THREADS * 16);
            asm volatile("global_load_async_to_lds_b128 %0, %1, off"
                         :: "v"(ldst), "v"(gsrc + it * THREADS * 4)
                         : "memory");
        }
        asm volatile("s_wait_asynccnt 0x0" ::: "memory");
    }
    __syncthreads();

    // ---- Demode in place: v_i = (|x_i - MODE| > THRESH) ? x_i : MODE ----
    for (int i = tid; i < KCHUNK; i += THREADS) {
        float xv = vsh[i];
        float xd = xv - MODE_C;
        vsh[i] = (fabsf(xd) > THRESH_C) ? xv : MODE_C;
    }
    __syncthreads();

    const int wave = tid >> 5;
    const int lane = tid & 31;
    const int half = lane >> 4;     // 0: K rows {0,1}, 1: K rows {2,3}
    const int n    = lane & 15;     // column within the 16-wide tile
    const int col  = blockIdx.x * COLS_PER_BLOCK + wave * 16 + n;

    const float* wp   = W + (size_t)k0 * OUT_F + col;
    const v2f*   vsh2 = (const v2f*)vsh;

    v8f acc = {0.f, 0.f, 0.f, 0.f, 0.f, 0.f, 0.f, 0.f};

    // One V_WMMA_F32_16X16X4_F32 per 4 K-rows x 16 columns of W.
    // A (16x4): VGPR0 = v[k+2*half], VGPR1 = v[k+1+2*half] (rows replicated)
    // B (4x16): VGPR0 = W[k+2*half][col], VGPR1 = W[k+1+2*half][col]
    #pragma unroll 8
    for (int k = 0; k < KCHUNK; k += 4) {
        v2f a = vsh2[(k >> 1) + half];               // ds_load_b64, 8B aligned

        const float* wrow = wp + (size_t)(k + 2 * half) * OUT_F;
        v2f b;
        b.x = __builtin_nontemporal_load(wrow);          // 64B sector / half-wave
        b.y = __builtin_nontemporal_load(wrow + OUT_F);  // next K row

        acc = __builtin_amdgcn_wmma_f32_16x16x4_f32(
            /*neg_a=*/false, a, /*neg_b=*/false, b,
            /*c_mod=*/(short)0, acc, /*reuse_a=*/false, /*reuse_b=*/false);
    }

    // D row 0 lives in VGPR0 of lanes 0..15 (M=0, N=lane).
    if (lane < 16) {
        partial[(size_t)blockIdx.y * OUT_F + col] = acc[0];
    }
}

__global__ __launch_bounds__(THREADS)
void scap_reduce_kernel(const float* __restrict__ partial,
                        const float* __restrict__ bias,
                        float* __restrict__ out) {
    int j = blockIdx.x * THREADS + threadIdx.x;
    if (j < OUT_F) {
        float s = bias[j];
        #pragma unroll
        for (int c = 0; c < KSPLIT; ++c)
            s += partial[(size_t)c * OUT_F + j];
        out[j] = s;
    }
}

extern "C" void kernel_launch(void* const* d_in, const int* in_sizes, int n_in,
                              void* d_out, int out_size, void* d_ws, size_t ws_size,
                              hipStream_t stream) {
    (void)in_sizes; (void)n_in; (void)out_size; (void)ws_size;

    const float* x    = (const float*)d_in[0];   // [1, 8192]
    const float* W    = (const float*)d_in[1];   // [8192, 28672] row-major
    const float* bias = (const float*)d_in[2];   // [28672]
    float* out        = (float*)d_out;           // [1, 28672]
    float* partial    = (float*)d_ws;            // KSPLIT * OUT_F floats (448 KB)

    dim3 grid(OUT_F / COLS_PER_BLOCK, KSPLIT);   // (224, 4) -> 7168 waves
    scap_gemv_wmma_kernel<<<grid, THREADS, 0, stream>>>(x, W, partial);

    dim3 rgrid((OUT_F + THREADS - 1) / THREADS); // 112 blocks
    scap_reduce_kernel<<<rgrid, THREADS, 0, stream>>>(partial, bias, out);
}